// BilinearAttention_9517647528306
// MI455X (gfx1250) — compile-verified
//
#include <hip/hip_runtime.h>

// CDNA5 / gfx1250 fp32 WMMA bilinear attention.
//
//   1) inter = Q @ W                               (WMMA f32 GEMM)
//   2) logits = mask>0 ? relu(inter @ V^T) : -1e9  (WMMA f32 GEMM + fused epilogue)
//   3) attn = softmax_rows(logits)
//   4) out  = attn @ V                             (WMMA f32 GEMM)
//
// GEMM: 128x128 block tile, 8 waves (4x2), 32x64 per wave, K-tile 16,
// double-buffered LDS fed by GLOBAL_LOAD_ASYNC_TO_LDS_B128 (ASYNCcnt),
// tiles stored [row][k] with 20-float stride so WMMA fragments are single
// conflict-free ds_load_b64's.
//
// d_ws must hold B*M*H floats (64 MB) for the intermediate.

typedef __attribute__((ext_vector_type(2))) float v2f;
typedef __attribute__((ext_vector_type(8))) float v8f;
typedef int v4i_gcc __attribute__((vector_size(16)));   // matches builtin pointee

#define MINVAL (-1e9f)

#if defined(__has_builtin)
#if __has_builtin(__builtin_amdgcn_global_load_async_to_lds_b128)
#define HAVE_ASYNC_LDS 1
#endif
#endif
#ifndef HAVE_ASYNC_LDS
#define HAVE_ASYNC_LDS 0
#endif

#if HAVE_ASYNC_LDS
__device__ __forceinline__ void async_cp16(float* lds, const float* gsrc) {
    __builtin_amdgcn_global_load_async_to_lds_b128(
        (__attribute__((address_space(1))) v4i_gcc*)gsrc,
        (__attribute__((address_space(3))) v4i_gcc*)lds,
        0, 0);
}
#define WAIT_ASYNC() asm volatile("s_wait_asynccnt 0x0" ::: "memory")
#else
#define WAIT_ASYNC()
#endif

// ---------------------------------------------------------------------------
// C[M,N] = A[M,K] * B[K,N].  BT=true: B stored N-major (B[k][n] = Bm[n*ldB+k]).
// EPI==1: fused relu + mask -> attention logits.
// ---------------------------------------------------------------------------
template <int EPI, bool BT>
__global__ __launch_bounds__(256) void gemm_f32_wmma(
    const float* __restrict__ A, const float* __restrict__ Bm,
    float* __restrict__ C, const int* __restrict__ mask,
    int K, int ldA, int ldB, int ldC, int maskLd,
    long long strideA, long long strideB, long long strideC, long long strideMask)
{
    constexpr int LDK = 20;                 // 16 k's + 4 pad (banks: phases disjoint)
    __shared__ float As[2][128 * LDK];
    __shared__ float Bs[2][128 * LDK];

    const int b = blockIdx.z;
    A  += (long long)b * strideA;
    Bm += (long long)b * strideB;
    C  += (long long)b * strideC;
    if (EPI) mask += (long long)b * strideMask;

    const int nBlock = blockIdx.x * 128;
    const int mBlock = blockIdx.y * 128;

    const int tid   = threadIdx.x;
    const int lane  = tid & 31;
    const int wave  = tid >> 5;
    const int waveM = wave >> 1;            // 0..3 -> 32 rows
    const int waveN = wave & 1;             // 0..1 -> 64 cols
    const int lm    = lane & 15;
    const int lh    = lane >> 4;            // K-phase select (ISA 32-bit A/B layout)

    // Staging coordinates: contiguous-16B chunks, [row][k] tile layout.
    const int r = tid & 127;                // tile row (m or n)
    const int q = tid >> 7;                 // which 8-k half
    const float* aSrc  = A  + (long long)(mBlock + r) * ldA + q * 8;   // + kBase
    const float* bSrcT = Bm + (long long)(nBlock + r) * ldB + q * 8;   // + kBase (BT)
    const int bkRow = tid >> 4;             // 0..15  (row-major B staging)
    const int bn0   = (tid & 15) * 8;
    const float* bSrcN = Bm + (long long)bkRow * ldB + nBlock + bn0;   // + kBase*ldB
    const int rdOff = r * LDK + q * 8;      // LDS chunk offset (16B aligned)

    v8f acc[2][4] = {};
    const int nt = K / 16;

    // ---- prologue: stage tile 0 into buffer 0 ----
    {
#if HAVE_ASYNC_LDS
        async_cp16(&As[0][rdOff],     aSrc);
        async_cp16(&As[0][rdOff + 4], aSrc + 4);
        if (BT) {
            async_cp16(&Bs[0][rdOff],     bSrcT);
            async_cp16(&Bs[0][rdOff + 4], bSrcT + 4);
        }
#else
        float4 a0 = *(const float4*)aSrc, a1 = *(const float4*)(aSrc + 4);
        *(float4*)&As[0][rdOff]     = a0;
        *(float4*)&As[0][rdOff + 4] = a1;
        if (BT) {
            float4 b0 = *(const float4*)bSrcT, b1 = *(const float4*)(bSrcT + 4);
            *(float4*)&Bs[0][rdOff]     = b0;
            *(float4*)&Bs[0][rdOff + 4] = b1;
        }
#endif
        if (!BT) {  // transpose-stage row-major B: [k][n] -> [n][k]
            float4 b0 = *(const float4*)bSrcN, b1 = *(const float4*)(bSrcN + 4);
#pragma unroll
            for (int j = 0; j < 4; ++j) {
                Bs[0][(bn0 + j) * LDK + bkRow]     = ((const float*)&b0)[j];
                Bs[0][(bn0 + 4 + j) * LDK + bkRow] = ((const float*)&b1)[j];
            }
        }
    }

    for (int kt = 0; kt < nt; ++kt) {
        const int cur = kt & 1;
        const int nxt = cur ^ 1;
        const bool hasNext = (kt + 1 < nt);

        WAIT_ASYNC();          // this wave's async loads for tile kt are in LDS
        __syncthreads();       // everyone's loads/stores for tile kt visible

        // ---- issue tile kt+1 (overlaps with compute below) ----
        float4 pb0, pb1;
        if (hasNext) {
            const int kB = (kt + 1) * 16;
#if HAVE_ASYNC_LDS
            async_cp16(&As[nxt][rdOff],     aSrc + kB);
            async_cp16(&As[nxt][rdOff + 4], aSrc + kB + 4);
            if (BT) {
                async_cp16(&Bs[nxt][rdOff],     bSrcT + kB);
                async_cp16(&Bs[nxt][rdOff + 4], bSrcT + kB + 4);
            }
#else
            float4 a0 = *(const float4*)(aSrc + kB), a1 = *(const float4*)(aSrc + kB + 4);
            *(float4*)&As[nxt][rdOff]     = a0;
            *(float4*)&As[nxt][rdOff + 4] = a1;
            if (BT) {
                float4 b0 = *(const float4*)(bSrcT + kB), b1 = *(const float4*)(bSrcT + kB + 4);
                *(float4*)&Bs[nxt][rdOff]     = b0;
                *(float4*)&Bs[nxt][rdOff + 4] = b1;
            }
#endif
            if (!BT) {  // loads now, LDS stores after compute (latency hidden)
                const float* s = bSrcN + (long long)kB * ldB;
                pb0 = *(const float4*)s;
                pb1 = *(const float4*)(s + 4);
            }
        }

        // ---- compute: 4 k-steps x 8 wmma tiles from buffer `cur` ----
#pragma unroll
        for (int kk0 = 0; kk0 < 16; kk0 += 4) {
            const int kk = kk0 + 2 * lh;
            v2f af[2], bf[4];
#pragma unroll
            for (int ti = 0; ti < 2; ++ti)
                af[ti] = *(const v2f*)&As[cur][(waveM * 32 + ti * 16 + lm) * LDK + kk];
#pragma unroll
            for (int tj = 0; tj < 4; ++tj)
                bf[tj] = *(const v2f*)&Bs[cur][(waveN * 64 + tj * 16 + lm) * LDK + kk];
#pragma unroll
            for (int tj = 0; tj < 4; ++tj)
#pragma unroll
                for (int ti = 0; ti < 2; ++ti)
                    acc[ti][tj] = __builtin_amdgcn_wmma_f32_16x16x4_f32(
                        false, af[ti], false, bf[tj],
                        (short)0, acc[ti][tj], false, false);
        }

        if (hasNext && !BT) {   // finish transpose-staging B for tile kt+1
#pragma unroll
            for (int j = 0; j < 4; ++j) {
                Bs[nxt][(bn0 + j) * LDK + bkRow]     = ((const float*)&pb0)[j];
                Bs[nxt][(bn0 + 4 + j) * LDK + bkRow] = ((const float*)&pb1)[j];
            }
        }
    }

    // ---- epilogue: C/D layout = VGPR r -> row r (lanes 0-15) / r+8 (16-31) ----
#pragma unroll
    for (int ti = 0; ti < 2; ++ti) {
        const int row0 = mBlock + waveM * 32 + ti * 16 + lh * 8;
#pragma unroll
        for (int tj = 0; tj < 4; ++tj) {
            const int col = nBlock + waveN * 64 + tj * 16 + lm;
#pragma unroll
            for (int rr = 0; rr < 8; ++rr) {
                const int row = row0 + rr;
                float v = acc[ti][tj][rr];
                if (EPI == 1) {
                    v = fmaxf(v, 0.0f);
                    v = (mask[(long long)row * maskLd + col] > 0) ? v : MINVAL;
                }
                C[(long long)row * ldC + col] = v;
            }
        }
    }
}

// ---------------------------------------------------------------------------
// Row softmax, one 256-thread block per row of length 2048 (8 elems/thread).
// ---------------------------------------------------------------------------
__global__ __launch_bounds__(256) void softmax_rows_2048(float* __restrict__ attn)
{
    __shared__ float red[256];
    float* p = attn + (long long)blockIdx.x * 2048;
    const int tid = threadIdx.x;

    float vals[8];
    float mx = -1e30f;
#pragma unroll
    for (int i = 0; i < 8; ++i) {
        vals[i] = p[tid + i * 256];
        mx = fmaxf(mx, vals[i]);
    }
    red[tid] = mx;
    __syncthreads();
    for (int s = 128; s > 0; s >>= 1) {
        if (tid < s) red[tid] = fmaxf(red[tid], red[tid + s]);
        __syncthreads();
    }
    mx = red[0];
    __syncthreads();

    float sum = 0.0f;
#pragma unroll
    for (int i = 0; i < 8; ++i) {
        vals[i] = __expf(vals[i] - mx);
        sum += vals[i];
    }
    red[tid] = sum;
    __syncthreads();
    for (int s = 128; s > 0; s >>= 1) {
        if (tid < s) red[tid] += red[tid + s];
        __syncthreads();
    }
    const float inv = 1.0f / red[0];
#pragma unroll
    for (int i = 0; i < 8; ++i) p[tid + i * 256] = vals[i] * inv;
}

// ---------------------------------------------------------------------------
extern "C" void kernel_launch(void* const* d_in, const int* in_sizes, int n_in,
                              void* d_out, int out_size, void* d_ws, size_t ws_size,
                              hipStream_t stream)
{
    (void)in_sizes; (void)n_in; (void)out_size; (void)ws_size;
    const int B = 8, M = 2048, N = 2048, H = 1024;

    const float* q    = (const float*)d_in[0];   // (B, M, H)
    const float* val  = (const float*)d_in[1];   // (B, N, H)
    const int*   mask = (const int*)d_in[2];     // (B, M, N)
    const float* W    = (const float*)d_in[3];   // (H, H)

    float* out   = (float*)d_out;                  // (B, M, H)
    float* attn  = out + (long long)B * M * H;     // (B, M, N)
    float* inter = (float*)d_ws;                   // (B, M, H) scratch

    dim3 blk(256);

    // 1) inter = Q @ W
    dim3 g1(H / 128, M / 128, B);
    gemm_f32_wmma<0, false><<<g1, blk, 0, stream>>>(
        q, W, inter, nullptr,
        /*K=*/H, /*ldA=*/H, /*ldB=*/H, /*ldC=*/H, /*maskLd=*/0,
        (long long)M * H, 0LL, (long long)M * H, 0LL);

    // 2) logits = mask>0 ? relu(inter @ V^T) : -1e9
    dim3 g2(N / 128, M / 128, B);
    gemm_f32_wmma<1, true><<<g2, blk, 0, stream>>>(
        inter, val, attn, mask,
        /*K=*/H, /*ldA=*/H, /*ldB=*/H, /*ldC=*/N, /*maskLd=*/N,
        (long long)M * H, (long long)N * H, (long long)M * N, (long long)M * N);

    // 3) attn = softmax(logits) in place
    softmax_rows_2048<<<B * M, blk, 0, stream>>>(attn);

    // 4) out = attn @ V
    dim3 g3(H / 128, M / 128, B);
    gemm_f32_wmma<0, false><<<g3, blk, 0, stream>>>(
        attn, val, out, nullptr,
        /*K=*/N, /*ldA=*/N, /*ldB=*/H, /*ldC=*/H, /*maskLd=*/0,
        (long long)M * N, (long long)N * H, (long long)M * H, 0LL);
}